// GCN_63943473103088
// MI455X (gfx1250) — compile-verified
//
#include <hip/hip_runtime.h>
#include <hip/hip_bf16.h>

typedef __attribute__((ext_vector_type(2))) float v2f;
typedef __attribute__((ext_vector_type(8))) float v8f;

#define F_IN      128
#define EMB       128
#define NCLS      64
#define NEG_SLOPE 0.01f

// ---------------------------------------------------------------------------
// Phase 1: degree (self-loop weight 1 + incoming edge weights), then rsqrt.
// ---------------------------------------------------------------------------
__global__ void init_deg_kernel(float* __restrict__ deg, int N) {
    int i = blockIdx.x * blockDim.x + threadIdx.x;
    if (i < N) deg[i] = 1.0f;  // self-loop contributes weight 1
}

__global__ void edge_deg_kernel(const long long* __restrict__ dst,
                                const float* __restrict__ w,
                                float* __restrict__ deg, int E) {
    int e = blockIdx.x * blockDim.x + threadIdx.x;
    if (e < E) atomicAdd(&deg[dst[e]], w[e]);
}

__global__ void dinv_kernel(float* __restrict__ deg, int N) {
    int i = blockIdx.x * blockDim.x + threadIdx.x;
    if (i < N) {
        float d = deg[i];
        deg[i] = (d > 0.0f) ? rsqrtf(d) : 0.0f;   // becomes dinv in-place
    }
}

// agg initialized with the conv bias so the scatter lands on top of it.
__global__ void init_agg_kernel(float* __restrict__ agg,
                                const float* __restrict__ b_conv,
                                long long total) {
    long long i = (long long)blockIdx.x * blockDim.x + threadIdx.x;
    if (i < total) agg[i] = b_conv[i & (EMB - 1)];
}

// ---------------------------------------------------------------------------
// Phase 2: h = x @ W_conv  via v_wmma_f32_16x16x4_f32.
// One wave computes a 16x16 tile of h; K stepped by 4 (32 WMMAs / tile).
// A layout (16x4 f32): lanes 0-15 -> K={0,1}, lanes 16-31 -> K={2,3}, M=lane&15
// B layout (4x16 f32): mirror of A with N=lane&15
// C/D layout: VGPR r -> row r (lanes 0-15) / row r+8 (lanes 16-31)
// ---------------------------------------------------------------------------
__global__ void gemm_xw_kernel(const float* __restrict__ x,
                               const float* __restrict__ W,
                               float* __restrict__ h, int N) {
    const int lane = threadIdx.x & 31;
    const int wave = threadIdx.x >> 5;
    const int tile = blockIdx.x * 8 + wave;
    const int tilesPerRow = EMB / 16;                 // 8
    const int numTiles = (N / 16) * tilesPerRow;
    if (tile >= numTiles) return;

    const int tm   = tile / tilesPerRow;
    const int tn   = tile % tilesPerRow;
    const int l15  = lane & 15;
    const int kOff = (lane >> 4) << 1;                // 0 or 2

    const float* xrow = x + (size_t)(tm * 16 + l15) * F_IN;
    const int    col  = tn * 16 + l15;

    v8f acc = {};
    for (int k = 0; k < F_IN; k += 4) {
        v2f a, b;
        a.x = xrow[k + kOff];
        a.y = xrow[k + kOff + 1];
        b.x = W[(size_t)(k + kOff)     * EMB + col];
        b.y = W[(size_t)(k + kOff + 1) * EMB + col];
        acc = __builtin_amdgcn_wmma_f32_16x16x4_f32(
                  false, a, false, b, (short)0, acc, false, false);
    }

    const int rBase = tm * 16 + ((lane >> 4) << 3);   // +8 for upper half-wave
#pragma unroll
    for (int r = 0; r < 8; ++r)
        h[(size_t)(rBase + r) * EMB + col] = acc[r];
}

// ---------------------------------------------------------------------------
// Phase 3: agg[dst] += dinv[src]*w*dinv[dst] * h[src]   (one wave per edge;
// lane holds a contiguous float4 slice -> fully coalesced gather + atomics).
// Edges e >= E are the N self-loops (src=dst=e-E, w=1).
// h and agg are ~25.6 MB each -> resident in MI455X's 192 MB L2.
// ---------------------------------------------------------------------------
__global__ void scatter_kernel(const long long* __restrict__ src,
                               const long long* __restrict__ dst,
                               const float* __restrict__ w,
                               const float* __restrict__ dinv,
                               const float* __restrict__ h,
                               float* __restrict__ agg, int E, int N) {
    const int lane = threadIdx.x & 31;
    const int e    = blockIdx.x * 8 + (threadIdx.x >> 5);
    if (e >= E + N) return;

    long long s, d;
    float ew;
    if (e < E) { s = src[e]; d = dst[e]; ew = w[e]; }
    else       { s = d = e - E;          ew = 1.0f; }

    const float coeff = dinv[s] * ew * dinv[d];
    const float4 v = ((const float4*)(h + (size_t)s * EMB))[lane];
    float* a = agg + (size_t)d * EMB + lane * 4;
    atomicAdd(a + 0, coeff * v.x);
    atomicAdd(a + 1, coeff * v.y);
    atomicAdd(a + 2, coeff * v.z);
    atomicAdd(a + 3, coeff * v.w);
}

// ---------------------------------------------------------------------------
// Phase 4: out = leaky_relu(agg) @ W_out + b_out  (leaky fused into A loads,
// bias fused into the store). Same WMMA tiling as phase 2.
// ---------------------------------------------------------------------------
__global__ void gemm_out_kernel(const float* __restrict__ agg,
                                const float* __restrict__ Wout,
                                const float* __restrict__ bout,
                                float* __restrict__ out, int N) {
    const int lane = threadIdx.x & 31;
    const int wave = threadIdx.x >> 5;
    const int tile = blockIdx.x * 8 + wave;
    const int tilesPerRow = NCLS / 16;                // 4
    const int numTiles = (N / 16) * tilesPerRow;
    if (tile >= numTiles) return;

    const int tm   = tile / tilesPerRow;
    const int tn   = tile % tilesPerRow;
    const int l15  = lane & 15;
    const int kOff = (lane >> 4) << 1;

    const float* arow = agg + (size_t)(tm * 16 + l15) * EMB;
    const int    col  = tn * 16 + l15;

    v8f acc = {};
    for (int k = 0; k < EMB; k += 4) {
        float a0 = arow[k + kOff];
        float a1 = arow[k + kOff + 1];
        a0 = (a0 >= 0.0f) ? a0 : NEG_SLOPE * a0;      // leaky_relu fused
        a1 = (a1 >= 0.0f) ? a1 : NEG_SLOPE * a1;
        v2f a, b;
        a.x = a0; a.y = a1;
        b.x = Wout[(size_t)(k + kOff)     * NCLS + col];
        b.y = Wout[(size_t)(k + kOff + 1) * NCLS + col];
        acc = __builtin_amdgcn_wmma_f32_16x16x4_f32(
                  false, a, false, b, (short)0, acc, false, false);
    }

    const float bias  = bout[col];
    const int   rBase = tm * 16 + ((lane >> 4) << 3);
#pragma unroll
    for (int r = 0; r < 8; ++r)
        out[(size_t)(rBase + r) * NCLS + col] = acc[r] + bias;
}

// ---------------------------------------------------------------------------
extern "C" void kernel_launch(void* const* d_in, const int* in_sizes, int n_in,
                              void* d_out, int out_size, void* d_ws, size_t ws_size,
                              hipStream_t stream) {
    (void)n_in; (void)out_size; (void)ws_size;

    const float*     x   = (const float*)d_in[0];
    const long long* ei  = (const long long*)d_in[1];  // int64 [2, E]
    const float*     ew  = (const float*)d_in[2];
    const float*     Wc  = (const float*)d_in[3];
    const float*     bc  = (const float*)d_in[4];
    const float*     Wo  = (const float*)d_in[5];
    const float*     bo  = (const float*)d_in[6];
    float*           out = (float*)d_out;

    const int E = in_sizes[2];
    const int N = in_sizes[0] / F_IN;
    const long long* src = ei;
    const long long* dst = ei + E;

    // workspace layout: [dinv N][h N*128][agg N*128], 16B-aligned pieces
    float* deg = (float*)d_ws;
    const size_t Npad = ((size_t)N + 63) & ~(size_t)63;
    float* h   = deg + Npad;
    float* agg = h + (size_t)N * EMB;

    // degree + normalization
    init_deg_kernel<<<(N + 255) / 256, 256, 0, stream>>>(deg, N);
    edge_deg_kernel<<<(E + 255) / 256, 256, 0, stream>>>(dst, ew, deg, E);
    dinv_kernel<<<(N + 255) / 256, 256, 0, stream>>>(deg, N);

    // agg <- bias
    const long long total = (long long)N * EMB;
    init_agg_kernel<<<(int)((total + 255) / 256), 256, 0, stream>>>(agg, bc, total);

    // h = x @ W_conv (fp32 WMMA)
    const int t1 = (N / 16) * (EMB / 16);
    gemm_xw_kernel<<<(t1 + 7) / 8, 256, 0, stream>>>(x, Wc, h, N);

    // edge scatter (L2-resident)
    const int ne = E + N;
    scatter_kernel<<<(ne + 7) / 8, 256, 0, stream>>>(src, dst, ew, deg, h, agg, E, N);

    // out = leaky(agg) @ W_out + b_out (fp32 WMMA)
    const int t2 = (N / 16) * (NCLS / 16);
    gemm_out_kernel<<<(t2 + 7) / 8, 256, 0, stream>>>(agg, Wo, bo, out, N);
}